// GraphConvolutionSparse_83528523973324
// MI455X (gfx1250) — compile-verified
//
#include <hip/hip_runtime.h>

#define NNODES 100000
#define NEDGES 1600000
#define D 64

typedef __attribute__((ext_vector_type(2))) float v2f;
typedef __attribute__((ext_vector_type(8))) float v8f;

// ---------------- Kernel 0: zero the aggregation scratch ----------------
__global__ void zero_ws_kernel(float* __restrict__ ws, int n4) {
    int i = blockIdx.x * blockDim.x + threadIdx.x;
    if (i < n4) {
        float4 z = make_float4(0.f, 0.f, 0.f, 0.f);
        reinterpret_cast<float4*>(ws)[i] = z;
    }
}

// ---------------- Kernel 1: edge scatter (gather * val, atomic segment sum) ----
// One wave (32 lanes) per edge. Each lane handles feature dims [2*lane, 2*lane+1]
// as a float2 load + two hardware fp32 global atomics into agg[row].
__global__ void edge_scatter_kernel(const float* __restrict__ feat,
                                    const int*   __restrict__ rows,
                                    const int*   __restrict__ cols,
                                    const float* __restrict__ vals,
                                    float*       __restrict__ agg) {
    unsigned gtid = blockIdx.x * blockDim.x + threadIdx.x;
    unsigned e    = gtid >> 5;           // wave index == edge index
    unsigned lane = threadIdx.x & 31u;
    if (e >= NEDGES) return;

    int   r = rows[e];
    int   c = cols[e];
    float v = vals[e];

    const float2 f = *reinterpret_cast<const float2*>(feat + (size_t)c * D + 2u * lane);
    float* dst = agg + (size_t)r * D + 2u * lane;
    // hardware global_atomic_add_f32 (no CAS loop)
    unsafeAtomicAdd(dst + 0, f.x * v);
    unsafeAtomicAdd(dst + 1, f.y * v);
}

// ---------------- Kernel 2: agg @ W + ReLU via V_WMMA_F32_16X16X4_F32 --------
// Block = 256 threads = 8 waves. Wave w computes output tile:
//   rows [blockIdx.x*32 + (w>>2)*16 , +16), cols [(w&3)*16, +16)
// K = 64 -> 16 WMMA(16x16x4) accumulation steps.
__global__ void __launch_bounds__(256)
gemm_relu_kernel(const float* __restrict__ agg,
                 const float* __restrict__ W,
                 float*       __restrict__ out) {
    __shared__ float lds_w[D * D];   // 16 KB

    const int tid = threadIdx.x;
    // cooperative load of W (4096 floats) into LDS as float4
    for (int i = tid; i < (D * D) / 4; i += 256) {
        reinterpret_cast<float4*>(lds_w)[i] =
            reinterpret_cast<const float4*>(W)[i];
    }
    __syncthreads();

    const int wave = tid >> 5;           // 0..7
    const int lane = tid & 31;
    const int m     = lane & 15;         // M (for A) / N (for B,C/D) within tile
    const int khalf = lane >> 4;         // 0 or 1 -> K sub-offset 0 or 2

    const int row0 = blockIdx.x * 32 + (wave >> 2) * 16;
    const int col0 = (wave & 3) * 16;

    const float* aRow = agg + (size_t)(row0 + m) * D;   // this lane's A row
    const float* bCol = lds_w + col0 + m;               // this lane's B column

    v8f acc = {};
    #pragma unroll
    for (int k = 0; k < 16; ++k) {
        const int kb = 4 * k + 2 * khalf;     // base K for this lane's A/B frags
        // A fragment: 16x4 f32 -> lane holds (M=m, K=kb) and (M=m, K=kb+1)
        float2 av = *reinterpret_cast<const float2*>(aRow + kb);
        v2f a; a.x = av.x; a.y = av.y;
        // B fragment: 4x16 f32 -> lane holds (K=kb, N=m) and (K=kb+1, N=m)
        v2f b;
        b.x = bCol[(size_t)kb * D];
        b.y = bCol[(size_t)(kb + 1) * D];
        acc = __builtin_amdgcn_wmma_f32_16x16x4_f32(
            /*neg_a=*/false, a, /*neg_b=*/false, b,
            /*c_mod=*/(short)0, acc, /*reuse_a=*/false, /*reuse_b=*/false);
    }

    // C/D layout: VGPR v -> M = v + 8*khalf, N = m
    float* oCol = out + col0 + m;
    #pragma unroll
    for (int v = 0; v < 8; ++v) {
        const int rr = row0 + v + 8 * khalf;
        float x = acc[v];
        oCol[(size_t)rr * D] = x > 0.f ? x : 0.f;
    }
}

// ----------------------------- launcher --------------------------------------
extern "C" void kernel_launch(void* const* d_in, const int* in_sizes, int n_in,
                              void* d_out, int out_size, void* d_ws, size_t ws_size,
                              hipStream_t stream) {
    const float* features = (const float*)d_in[0];   // [NNODES*D]
    const int*   edge_rows = (const int*)d_in[1];    // [NEDGES]
    const int*   edge_cols = (const int*)d_in[2];    // [NEDGES]
    const float* edge_vals = (const float*)d_in[3];  // [NEDGES]
    const float* weight    = (const float*)d_in[4];  // [D*D]
    float* out = (float*)d_out;                      // [NNODES*D]
    float* agg = (float*)d_ws;                       // [NNODES*D] scratch

    // 0) zero agg (NNODES*D floats, as float4)
    {
        int n4 = (NNODES * D) / 4;                   // 1,600,000
        int blocks = (n4 + 255) / 256;
        zero_ws_kernel<<<blocks, 256, 0, stream>>>(agg, n4);
    }
    // 1) scatter: one wave per edge -> NEDGES*32 threads
    {
        long long threads = (long long)NEDGES * 32;
        int blocks = (int)((threads + 255) / 256);   // 200,000
        edge_scatter_kernel<<<blocks, 256, 0, stream>>>(
            features, edge_rows, edge_cols, edge_vals, agg);
    }
    // 2) WMMA GEMM + ReLU: block covers 32 rows x 64 cols; 100000/32 = 3125
    {
        int blocks = NNODES / 32;                    // 3125, exact
        gemm_relu_kernel<<<blocks, 256, 0, stream>>>(agg, weight, out);
    }
}